// NeuralMJD_23888608101187
// MI455X (gfx1250) — compile-verified
//
#include <hip/hip_runtime.h>
#include <stdint.h>

// Problem constants (from reference setup_inputs)
#define B_   8
#define N_   500
#define T_   12
#define R_   300
#define H_   150
#define MAXN 5
#define LOG_2PI_HALF 0.9189385332046727f

typedef __attribute__((ext_vector_type(2))) float v2f;
typedef __attribute__((ext_vector_type(4))) float f4;
typedef __attribute__((ext_vector_type(4))) int   v4i;
typedef __attribute__((ext_vector_type(8))) float v8f;

// CDNA5 async global->LDS path (ASYNCcnt), with safe fallback
#if defined(__has_builtin)
#  if __has_builtin(__builtin_amdgcn_global_load_async_to_lds_b128) && \
      __has_builtin(__builtin_amdgcn_s_wait_asynccnt)
#    define USE_ASYNC_LDS 1
#  endif
#endif

// ---------------- counter-based RNG helpers ----------------
__device__ __forceinline__ uint32_t hash_u32(uint64_t x) {
  x ^= x >> 33; x *= 0xff51afd7ed558ccdULL;
  x ^= x >> 33; x *= 0xc4ceb9fe1a85ec53ULL;
  x ^= x >> 33;
  return (uint32_t)x;
}
__device__ __forceinline__ float u32_to_uniform(uint32_t u) {
  return (float)(u >> 8) * (1.0f / 16777216.0f) + (0.5f / 16777216.0f); // (0,1)
}
__device__ __forceinline__ float poisson_from_u(float u, float lam) {
  // inversion sampling; lam <= 0.1 here so k rarely exceeds 2
  float p   = __expf(-lam);
  float cdf = p;
  float k   = 0.0f;
#pragma unroll
  for (int i = 1; i <= 8; ++i) {
    if (u > cdf) { p *= lam / (float)i; cdf += p; k = (float)i; }
  }
  return k;
}
__device__ __forceinline__ float clipf(float x, float lo, float hi) {
  return fminf(fmaxf(x, lo), hi);
}

// ================= Kernel 1: MC jump-diffusion simulation =================
// grid = B*N blocks, block = 320 threads (10 wave32), thread r = MC run.
// Parameter prepare done once per block (thread 0 -> LDS).
// One Box-Muller pair per substep supplies BOTH diffusion (cos) and jump (sin)
// normals, halving transcendental work on the TRANS pipe.
__global__ __launch_bounds__(320) void mjd_simulate(
    const float* __restrict__ params,   // [B,N,T,5]
    const float* __restrict__ past,     // [B,N,4,8]
    const float* __restrict__ coef_,    // [B,N,1]
    const float* __restrict__ mn_,      // [B,N,1]
    const int* __restrict__ pJ, const int* __restrict__ pRestart,
    float* __restrict__ s_out)          // [B,N,R,T]  de-normalized paths
{
  const int bn = blockIdx.x;
  const int r  = threadIdx.x;

  const int   J    = pJ[0];
  const int   restart = pRestart[0];
  const float invJ = 1.0f / (float)J;
  const float isqJ = 1.0f / __builtin_sqrtf((float)J);

  __shared__ float s_alpha[T_], s_pm[T_], s_lam[T_], s_nu[T_], s_asig[T_], s_agam[T_];
  __shared__ float s_ls0;

  if (threadIdx.x == 0) {
    const float* pp = params + (size_t)bn * T_ * 5;
    __builtin_prefetch(pp, 0, 0);   // global_prefetch_b8
    const float s0 = fmaxf(past[(size_t)bn * 32 + 7], 1e-6f);
    float cum = __logf(s0);
    s_ls0 = cum;
    for (int t = 0; t < T_; ++t) {
      float m = clipf(pp[t * 5 + 0], -10.0f, 10.0f);
      float s = clipf(pp[t * 5 + 1], 0.0f, 1.0f);
      float l = clipf(__expf(fminf(pp[t * 5 + 2], 0.0f)), 1e-6f, 1.0f);
      float n = clipf(pp[t * 5 + 3], -0.5f, 0.5f);
      float g = clipf(pp[t * 5 + 4], 0.0f, 1.0f);
      float k = __expf(n + 0.5f * g * g) - 1.0f;
      s_alpha[t] = (m - l * k - 0.5f * s * s) * invJ;
      s_pm[t]    = cum;                // prev_mean[t]
      s_lam[t]   = l;
      s_nu[t]    = n;
      s_asig[t]  = fabsf(s);
      s_agam[t]  = fabsf(g);
      cum += m;
    }
  }
  __syncthreads();
  if (r >= R_) return;

  const float coef = coef_[bn];
  const float mn   = mn_[bn];
  const float sgn  = (r < H_) ? 1.0f : -1.0f;   // antithetic variates
  const int   rH   = (r < H_) ? r : r - H_;
  const uint64_t bn64 = (uint64_t)bn;
  const float log_s0 = s_ls0;

  float log_s = log_s0;
  for (int t = 0; t < T_; ++t) {
    const float lam_sub = s_lam[t] * invJ;
    const float asig    = s_asig[t];
    const float agam    = s_agam[t];
    const float alpha   = s_alpha[t];
    const float nu      = s_nu[t];
    const float pm      = s_pm[t];
    float out_ls = log_s;
    for (int j = 0; j < J; ++j) {
      const uint64_t sid = (uint64_t)(t * J + j);
      const uint64_t cn  = ((bn64 * 4096ull + sid) * 2ull + 0ull) * 512ull + (uint64_t)rH;
      const uint64_t cp  = ((bn64 * 4096ull + sid) * 2ull + 1ull) * 512ull + (uint64_t)r;
      // one Box-Muller pair -> two independent normals (diffusion + jump)
      float u1 = u32_to_uniform(hash_u32(cn * 2ull + 0ull));
      float u2 = u32_to_uniform(hash_u32(cn * 2ull + 1ull));
      float rad = __builtin_sqrtf(-2.0f * __logf(u1));
      float sn, cs;
      __sincosf(6.28318530718f * u2, &sn, &cs);
      float eb = sgn * rad * cs;
      float ez = sgn * rad * sn;
      float up = u32_to_uniform(hash_u32(cp * 2ull));
      float kc = poisson_from_u(up, lam_sub);
      float delta = alpha + asig * eb * isqJ
                  + kc * nu + __builtin_sqrtf(kc) * agam * ez;
      log_s += delta;
      if (j == J - 1) out_ls = log_s;
      if (j == 0 && restart && t > 0) log_s = pm + delta;
    }
    s_out[((size_t)bn * R_ + r) * T_ + t] = __expf(out_ls) * coef + mn;
  }
}

// ================= Kernel 2: winner selection via f32 WMMA =================
// grid = B*N blocks, block = 384 threads = 12 wave32, wave w handles t = w.
// The [R,T] path tile (14.4 KB) is loaded ONCE per block into LDS -- via the
// CDNA5 async-load-to-LDS path when available -- then shared by all 12 waves.
// lg(x,k) = u_k + v_k*x + w_k*x^2  -->  (6x3 coeffs) x (3x16 powers) per chunk,
// mapped to V_WMMA_F32_16X16X4_F32; components on the M axis so each lane 0-15
// holds all 6 component densities for its run in C VGPRs 0..5 (no shuffles).
__global__ __launch_bounds__(384) void mjd_select(
    const float* __restrict__ params, const float* __restrict__ past,
    const float* __restrict__ coef_,  const float* __restrict__ mn_,
    const float* __restrict__ target, const float* __restrict__ s_out,
    float* __restrict__ s_win, float* __restrict__ s_prob)
{
  const int bn   = blockIdx.x;
  const int t    = threadIdx.x >> 5;    // wave id = time step, 0..11
  const int lane = threadIdx.x & 31;

  __shared__ float tile[R_ * T_];       // [run][t], same layout as s_out slice

  const float* src = s_out + (size_t)bn * (R_ * T_);
#ifdef USE_ASYNC_LDS
  {
    v4i* g = (v4i*)src;    // builtin expects non-const generic int4*
    v4i* l = (v4i*)tile;
    for (int i = threadIdx.x; i < (R_ * T_) / 4; i += 384)
      __builtin_amdgcn_global_load_async_to_lds_b128(g + i, l + i, 0, 0);
    __builtin_amdgcn_s_wait_asynccnt(0);
  }
#else
  for (int i = threadIdx.x; i < (R_ * T_) / 4; i += 384)
    ((f4*)tile)[i] = ((const f4*)src)[i];
#endif
  __syncthreads();

  const float* pp = params + (size_t)bn * T_ * 5;
  __builtin_prefetch(pp, 0, 0);

  // prepare up to t: cum -> log_s_pred_mean[t] = prev_mean[t] + mus[t]
  float cum = __logf(fmaxf(past[(size_t)bn * 32 + 7], 1e-6f));
  float s_t = 0.0f, l_t = 1e-6f, n_t = 0.0f, g_t = 0.0f;
  for (int tt = 0; tt <= t; ++tt) {
    cum += clipf(pp[tt * 5 + 0], -10.0f, 10.0f);
    if (tt == t) {
      s_t = clipf(pp[tt * 5 + 1], 0.0f, 1.0f);
      l_t = clipf(__expf(fminf(pp[tt * 5 + 2], 0.0f)), 1e-6f, 1.0f);
      n_t = clipf(pp[tt * 5 + 3], -0.5f, 0.5f);
      g_t = clipf(pp[tt * 5 + 4], 0.0f, 1.0f);
    }
  }
  const float kq     = __expf(n_t + 0.5f * g_t * g_t) - 1.0f;
  const float a_base = cum - l_t * kq - 0.5f * s_t * s_t;

  // Per-component polynomial coefficients (component index = lane & 15)
  const int   kcomp = lane & 15;
  const bool  hi    = lane >= 16;
  const bool  vc    = kcomp <= MAXN;
  const float nk  = (float)kcomp;
  const float ak  = a_base + nk * n_t;
  const float bk2 = fmaxf(s_t * s_t + nk * g_t * g_t, 1e-6f);
  const float wk  = -0.5f / bk2;
  const float vk  = ak / bk2;
  const float uk  = -0.5f * __logf(bk2) - ak * ak * (0.5f / bk2) - LOG_2PI_HALF;

  // A fragment (16x4 f32): lanes 0-15 -> {K0=u, K1=v}; lanes 16-31 -> {K2=w, K3=0}
  v2f afrag;
  afrag.x = hi ? (vc ? wk : 0.0f) : (vc ? uk : 0.0f);
  afrag.y = hi ? 0.0f : (vc ? vk : 0.0f);

  const float lgam_tab[6] = {0.0f, 0.0f, 0.69314718f, 1.79175947f,
                             3.17805383f, 4.78749174f};
  const float loglam = __logf(l_t);
  const float coef   = coef_[bn];
  const float mn     = mn_[bn];
  const float tgt    = target[(size_t)bn * T_ + t];

  float best_lp  = -3.4e38f; int best_ip = 0;
  float best_err =  3.4e38f; int best_iw = 0;

  const int nchunk = (R_ + 15) / 16;   // 19
  for (int c = 0; c < nchunk; ++c) {
    const int  run  = c * 16 + (lane & 15);
    const bool vr   = run < R_;
    const int  ridx = vr ? run : 0;
    const float sdm = tile[ridx * T_ + t];
    const float x   = __logf(fmaxf((sdm - mn) / coef, 1e-30f)); // recover log_s

    // B fragment (4x16 f32): lanes 0-15 -> {K0=1, K1=x}; lanes 16-31 -> {K2=x^2, K3=0}
    v2f bfrag;
    bfrag.x = hi ? x * x : 1.0f;
    bfrag.y = hi ? 0.0f  : x;

    v8f cfrag = {};
    cfrag = __builtin_amdgcn_wmma_f32_16x16x4_f32(
        /*neg_a=*/false, afrag, /*neg_b=*/false, bfrag,
        /*c_mod=*/(short)0, cfrag, /*reuse_a=*/false, /*reuse_b=*/false);

    // lanes 0-15: cfrag[k] = gaussian log-density of component k for this run
    float sk[6];
#pragma unroll
    for (int k = 0; k < 6; ++k)
      sk[k] = cfrag[k] + (-l_t + loglam * (float)k - lgam_tab[k]);
    float mx = sk[0];
#pragma unroll
    for (int k = 1; k < 6; ++k) mx = fmaxf(mx, sk[k]);
    float se = 0.0f;
#pragma unroll
    for (int k = 0; k < 6; ++k) se += __expf(sk[k] - mx);
    float lp  = mx + __logf(se);
    float err = fabsf(sdm - tgt);
    if (!vr || hi) { lp = -3.4e38f; err = 3.4e38f; }
    if (lp  > best_lp)  { best_lp = lp;  best_ip = run; }
    if (err < best_err) { best_err = err; best_iw = run; }
  }

  // wave32 argmax / argmin reduction (per wave; hi lanes carry sentinels)
  for (int off = 16; off > 0; off >>= 1) {
    float olp = __shfl_down(best_lp, off, 32);
    int   oip = __shfl_down(best_ip, off, 32);
    float oer = __shfl_down(best_err, off, 32);
    int   oiw = __shfl_down(best_iw, off, 32);
    if (olp > best_lp)  { best_lp = olp;  best_ip = oip; }
    if (oer < best_err) { best_err = oer; best_iw = oiw; }
  }
  if (lane == 0) {
    const int gid = bn * T_ + t;
    s_win[gid]  = tile[best_iw * T_ + t];
    s_prob[gid] = tile[best_ip * T_ + t];
  }
}

// ================= host launcher =================
extern "C" void kernel_launch(void* const* d_in, const int* in_sizes, int n_in,
                              void* d_out, int out_size, void* d_ws, size_t ws_size,
                              hipStream_t stream) {
  (void)in_sizes; (void)n_in; (void)d_ws; (void)ws_size; (void)out_size;
  const float* params  = (const float*)d_in[0];
  const float* past    = (const float*)d_in[1];
  const float* coef    = (const float*)d_in[2];
  const float* mn      = (const float*)d_in[3];
  const float* target  = (const float*)d_in[4];
  const int*   pJ      = (const int*)d_in[5];
  const int*   pRest   = (const int*)d_in[6];

  float* out    = (float*)d_out;
  float* s_out  = out;                                       // [B,N,R,T]
  float* s_win  = out + (size_t)B_ * N_ * R_ * T_;           // [B,N,T]
  float* s_prob = s_win + (size_t)B_ * N_ * T_;              // [B,N,T]

  mjd_simulate<<<B_ * N_, 320, 0, stream>>>(params, past, coef, mn, pJ, pRest, s_out);
  mjd_select<<<B_ * N_, 384, 0, stream>>>(params, past, coef, mn, target,
                                          s_out, s_win, s_prob);
}